// ParticleGraphNetwork_68066641707592
// MI455X (gfx1250) — compile-verified
//
#include <hip/hip_runtime.h>
#include <hip/hip_bf16.h>

// ParticleGraphNetwork on gfx1250 (MI455X).
//
// Roofline: ~8.5 GFLOP total vs ~GBs of L2-resident traffic + 154M f32
// atomics -> memory/atomic bound. Hidden state (12.8 MB) fits in 192 MB L2.
// Edge MLP + predictor are factorized into per-node dot products (linear
// before sigmoid), eliminating the h[dst] gather and E x 128 MACs.
// GRU GEMMs use V_WMMA_F32_16X16X4_F32 (fp32 matches reference numerics;
// FLOPs are not the bottleneck so fp32 WMMA rate is fine).

typedef __attribute__((ext_vector_type(2))) float v2f;
typedef __attribute__((ext_vector_type(8))) float v8f;

__device__ __forceinline__ float sigmoidf_(float x) {
    return 1.0f / (1.0f + __expf(-x));
}

// ---------------------------------------------------------------- input linear
// h[n][j] = sum_k features[n][k] * W_in[k][j] + b_in[j]   (IN=16, H=64)
__global__ void pgn_input_linear(const float* __restrict__ feat,
                                 const float* __restrict__ W_in,
                                 const float* __restrict__ b_in,
                                 float* __restrict__ h,
                                 int n_nodes) {
    int idx = blockIdx.x * blockDim.x + threadIdx.x;
    if (idx >= n_nodes * 64) return;
    int n = idx >> 6;
    int j = idx & 63;
    const float* fr = feat + (size_t)n * 16;
    float acc = b_in[j];
#pragma unroll
    for (int k = 0; k < 16; ++k) acc += fr[k] * W_in[k * 64 + j];
    h[idx] = acc;
}

// ------------------------------------------------------------- node projection
// pu[n] = h[n] . Wvec[0:64] ; pv[n] = h[n] . Wvec[64:128]
// One wave32 per node; lane handles 2 features; shuffle-reduce.
__global__ void pgn_node_proj(const float* __restrict__ h,
                              const float* __restrict__ Wvec,
                              float* __restrict__ pu,
                              float* __restrict__ pv,
                              int n_nodes) {
    int wid  = (blockIdx.x * blockDim.x + threadIdx.x) >> 5;
    int lane = threadIdx.x & 31;
    if (wid >= n_nodes) return;
    v2f hv = *(const v2f*)(h + (size_t)wid * 64 + lane * 2);
    float su = hv.x * Wvec[lane * 2]      + hv.y * Wvec[lane * 2 + 1];
    float sv = hv.x * Wvec[64 + lane * 2] + hv.y * Wvec[64 + lane * 2 + 1];
#pragma unroll
    for (int off = 16; off > 0; off >>= 1) {
        su += __shfl_xor(su, off, 32);
        sv += __shfl_xor(sv, off, 32);
    }
    if (lane == 0) { pu[wid] = su; pv[wid] = sv; }
}

// ------------------------------------------------------------- edge -> scatter
// w = sigmoid(pu[src] + pv[dst] + b_edge); h_new[dst] += h[src] * w
// One wave32 per edge; lane handles 2 features; 2 f32 atomics per lane.
__global__ void pgn_edge_scatter(const float* __restrict__ h,
                                 const int* __restrict__ src,
                                 const int* __restrict__ dst,
                                 const float* __restrict__ pu,
                                 const float* __restrict__ pv,
                                 const float* __restrict__ b_edge,
                                 float* __restrict__ h_new,
                                 int n_edges) {
    int wid  = (blockIdx.x * blockDim.x + threadIdx.x) >> 5;
    int lane = threadIdx.x & 31;
    if (wid >= n_edges) return;
    int s = src[wid];
    int d = dst[wid];
    float w = sigmoidf_(pu[s] + pv[d] + b_edge[0]);
    v2f hv = *(const v2f*)(h + (size_t)s * 64 + lane * 2);
    float* outp = h_new + (size_t)d * 64 + lane * 2;
    atomicAdd(outp,     hv.x * w);
    atomicAdd(outp + 1, hv.y * w);
}

// ---------------------------------------------------------------- GRU via WMMA
// One wave32 per 16-node tile. For each 16-wide output-feature chunk c,
// accumulate 6 WMMA tiles (i_r,i_z,i_n from h_new@Wih^T; h_r,h_z,h_n from
// h@Whh^T) over K=64 with V_WMMA_F32_16X16X4_F32, then fused gate math.
//
// fp32 WMMA layouts (ISA 7.12.2):
//   A 16x4:  lanes 0-15 -> M=lane, VGPR{0,1} = K{0,1}; lanes 16-31 -> K{2,3}
//   B 4x16:  lanes 0-15 -> N=lane, VGPR{0,1} = K{0,1}; lanes 16-31 -> K{2,3}
//   C/D 16x16: VGPR r -> rows {r, r+8} split across lane halves, N = lane&15
__global__ void pgn_gru_wmma(const float* __restrict__ h_new,
                             const float* __restrict__ h_old,
                             float* __restrict__ h_out,
                             const float* __restrict__ Wih,   // [192][64]
                             const float* __restrict__ Whh,   // [192][64]
                             const float* __restrict__ bih,   // [192]
                             const float* __restrict__ bhh,   // [192]
                             int n_nodes) {
    int wave = (blockIdx.x * blockDim.x + threadIdx.x) >> 5;
    int lane = threadIdx.x & 31;
    int node_base = wave * 16;
    if (node_base >= n_nodes) return;

    int m    = lane & 15;
    int koff = (lane >> 4) << 1;   // 0 or 2
    int rclamp = node_base + m;
    if (rclamp > n_nodes - 1) rclamp = n_nodes - 1;

    // A fragments for the whole K=64 strip (16 k-steps of K=4)
    v2f a_new[16], a_old[16];
    const float* rowN = h_new + (size_t)rclamp * 64;
    const float* rowO = h_old + (size_t)rclamp * 64;
#pragma unroll
    for (int k = 0; k < 16; ++k) {
        int c0 = k * 4 + koff;
        a_new[k] = *(const v2f*)(rowN + c0);
        a_old[k] = *(const v2f*)(rowO + c0);
    }

#pragma unroll
    for (int c = 0; c < 4; ++c) {
        v8f ir = {}, iz = {}, inn = {}, hr = {}, hz = {}, hnn = {};
        int col = c * 16 + (lane & 15);    // output feature 0..63 (per lane N)
#pragma unroll
        for (int k = 0; k < 16; ++k) {
            int kb = k * 4 + koff;
            // B fragment rows: gate r -> Wih rows [0,64), z -> [64,128), n -> [128,192)
            v2f bri = *(const v2f*)(Wih + (size_t)(col      ) * 64 + kb);
            v2f bzi = *(const v2f*)(Wih + (size_t)(col +  64) * 64 + kb);
            v2f bni = *(const v2f*)(Wih + (size_t)(col + 128) * 64 + kb);
            v2f brh = *(const v2f*)(Whh + (size_t)(col      ) * 64 + kb);
            v2f bzh = *(const v2f*)(Whh + (size_t)(col +  64) * 64 + kb);
            v2f bnh = *(const v2f*)(Whh + (size_t)(col + 128) * 64 + kb);
            ir  = __builtin_amdgcn_wmma_f32_16x16x4_f32(false, a_new[k], false, bri, (short)0, ir,  false, false);
            iz  = __builtin_amdgcn_wmma_f32_16x16x4_f32(false, a_new[k], false, bzi, (short)0, iz,  false, false);
            inn = __builtin_amdgcn_wmma_f32_16x16x4_f32(false, a_new[k], false, bni, (short)0, inn, false, false);
            hr  = __builtin_amdgcn_wmma_f32_16x16x4_f32(false, a_old[k], false, brh, (short)0, hr,  false, false);
            hz  = __builtin_amdgcn_wmma_f32_16x16x4_f32(false, a_old[k], false, bzh, (short)0, hz,  false, false);
            hnn = __builtin_amdgcn_wmma_f32_16x16x4_f32(false, a_old[k], false, bnh, (short)0, hnn, false, false);
        }
        float b_ir = bih[col], b_iz = bih[64 + col], b_in = bih[128 + col];
        float b_hr = bhh[col], b_hz = bhh[64 + col], b_hn = bhh[128 + col];
        int rowadd = (lane >> 4) << 3;   // lanes 16-31 hold rows M+8
#pragma unroll
        for (int r8 = 0; r8 < 8; ++r8) {
            int row = node_base + r8 + rowadd;
            if (row >= n_nodes) continue;
            float rg = sigmoidf_(ir[r8] + b_ir + hr[r8] + b_hr);
            float zg = sigmoidf_(iz[r8] + b_iz + hz[r8] + b_hz);
            float ng = tanhf(inn[r8] + b_in + rg * (hnn[r8] + b_hn));
            float ho = h_old[(size_t)row * 64 + col];
            h_out[(size_t)row * 64 + col] = (1.0f - zg) * ng + zg * ho;
        }
    }
}

// ------------------------------------------------------------- edge prediction
// out[e] = qu[src[e]] + qv[dst[e]] + b_pred
__global__ void pgn_edge_out(const int* __restrict__ src,
                             const int* __restrict__ dst,
                             const float* __restrict__ qu,
                             const float* __restrict__ qv,
                             const float* __restrict__ b_pred,
                             float* __restrict__ out,
                             int n_edges) {
    int e = blockIdx.x * blockDim.x + threadIdx.x;
    if (e >= n_edges) return;
    out[e] = qu[src[e]] + qv[dst[e]] + b_pred[0];
}

extern "C" void kernel_launch(void* const* d_in, const int* in_sizes, int n_in,
                              void* d_out, int out_size, void* d_ws, size_t ws_size,
                              hipStream_t stream) {
    const float* features = (const float*)d_in[0];
    const int*   src      = (const int*)d_in[1];
    const int*   dst      = (const int*)d_in[2];
    const float* W_in     = (const float*)d_in[3];
    const float* b_in     = (const float*)d_in[4];
    const float* W_edge   = (const float*)d_in[5];
    const float* b_edge   = (const float*)d_in[6];
    const float* Wih      = (const float*)d_in[7];
    const float* Whh      = (const float*)d_in[8];
    const float* bih      = (const float*)d_in[9];
    const float* bhh      = (const float*)d_in[10];
    const float* W_pred   = (const float*)d_in[11];
    const float* b_pred   = (const float*)d_in[12];

    const int N = in_sizes[0] / 16;           // 50000
    const int E = in_sizes[1];                // 800000
    const int L = in_sizes[7] / (192 * 64);   // 3

    // Workspace carve-out: 3*N*64 + 2*N floats (~39 MB for N=50000)
    float* hA   = (float*)d_ws;
    float* hB   = hA + (size_t)N * 64;
    float* hNew = hB + (size_t)N * 64;
    float* pu   = hNew + (size_t)N * 64;
    float* pv   = pu + N;

    {
        int total = N * 64;
        pgn_input_linear<<<(total + 255) / 256, 256, 0, stream>>>(features, W_in, b_in, hA, N);
    }

    float* h_cur = hA;
    float* h_nxt = hB;
    for (int l = 0; l < L; ++l) {
        pgn_node_proj<<<(N + 7) / 8, 256, 0, stream>>>(h_cur, W_edge, pu, pv, N);
        hipMemsetAsync(hNew, 0, (size_t)N * 64 * sizeof(float), stream);
        pgn_edge_scatter<<<(E + 7) / 8, 256, 0, stream>>>(h_cur, src, dst, pu, pv, b_edge, hNew, E);
        int tiles = (N + 15) / 16;
        pgn_gru_wmma<<<(tiles + 7) / 8, 256, 0, stream>>>(
            hNew, h_cur, h_nxt,
            Wih + (size_t)l * 192 * 64, Whh + (size_t)l * 192 * 64,
            bih + (size_t)l * 192, bhh + (size_t)l * 192, N);
        float* t = h_cur; h_cur = h_nxt; h_nxt = t;
    }

    pgn_node_proj<<<(N + 7) / 8, 256, 0, stream>>>(h_cur, W_pred, pu, pv, N);
    pgn_edge_out<<<(E + 255) / 256, 256, 0, stream>>>(src, dst, pu, pv, b_pred, (float*)d_out, E);
}